// GroupNormNN_57638461112380
// MI455X (gfx1250) — compile-verified
//
#include <hip/hip_runtime.h>
#include <cstdint>

#define TPB 256

// ---------------------------------------------------------------------------
// CDNA5 Tensor Data Mover path (gfx1250 only). 6-arg builtin form:
//   (uint32x4 g0, int32x8 g1, int32x4 g2, int32x4 g3, int32x8 g4, i32 cpol)
// ---------------------------------------------------------------------------
#if defined(__gfx1250__) && __has_builtin(__builtin_amdgcn_tensor_load_to_lds)
#define USE_TDM 1
typedef unsigned int u32x4 __attribute__((ext_vector_type(4)));
typedef int          i32x8 __attribute__((ext_vector_type(8)));
typedef int          i32x4 __attribute__((ext_vector_type(4)));
#else
#define USE_TDM 0
#endif

typedef float f32x4 __attribute__((ext_vector_type(4)));

__device__ __forceinline__ void wg_wait_tensorcnt0() {
#if defined(__gfx1250__) && __has_builtin(__builtin_amdgcn_s_wait_tensorcnt)
  __builtin_amdgcn_s_wait_tensorcnt(0);
#endif
}
__device__ __forceinline__ void wg_wait_tensorcnt1() {
#if defined(__gfx1250__) && __has_builtin(__builtin_amdgcn_s_wait_tensorcnt)
  __builtin_amdgcn_s_wait_tensorcnt(1);
#endif
}

#if USE_TDM
// DMA a 96x96 tile of f32 (row pitch 512 elements in global) into LDS at
// lds_byte_addr (contiguous 96-float rows). tensor_d0/d1 = remaining extent
// from the tile origin, so hardware OOB handling zero-fills pad rows/cols.
__device__ __forceinline__ void tdm_load_tile96(const float* gsrc,
                                                unsigned lds_byte_addr,
                                                int tensor_d0, int tensor_d1) {
  unsigned long long ga = (unsigned long long)(uintptr_t)gsrc;
  u32x4 g0;
  g0.x = 1u;                                                  // count=1 (valid user D#)
  g0.y = lds_byte_addr;                                       // lds_addr [63:32]
  g0.z = (unsigned)(ga & 0xffffffffull);                      // global_addr[31:0]
  g0.w = (unsigned)((ga >> 32) & 0x1ffffffull) | (2u << 30);  // addr[56:32] | type=2
  i32x8 g1;
  g1.s0 = (int)(2u << 16);                                    // data_size=2 -> 4 bytes
  g1.s1 = (int)(((unsigned)tensor_d0 & 0xffffu) << 16);       // tensor_dim0[15:0]
  g1.s2 = (int)((((unsigned)tensor_d0 >> 16) & 0xffffu) |
                (((unsigned)tensor_d1 & 0xffffu) << 16));     // td0[31:16] | td1[15:0]
  g1.s3 = (int)((((unsigned)tensor_d1 >> 16) & 0xffffu) |
                (96u << 16));                                 // td1[31:16] | tile_dim0=96
  g1.s4 = 96;                                                 // tile_dim1=96, tile_dim2=0
  g1.s5 = 512;                                                // tensor_dim0_stride lo32
  g1.s6 = 0;                                                  // stride0 hi16 | stride1[15:0]
  g1.s7 = 4;                                                  // stride1[47:16] (512*512>>16)
  i32x4 gz4 = {0, 0, 0, 0};
  i32x8 gz8 = {0, 0, 0, 0, 0, 0, 0, 0};
  __builtin_amdgcn_tensor_load_to_lds(g0, g1, gz4, gz4, gz8, 0);
}
#endif

// ---------------------------------------------------------------------------
// One workgroup handles one (n, group, 64x64 output tile).
// Dynamic LDS layout (floats), 36864 total = 144 KB:
//   [    0, 9216) xb0        -- TDM dest buffer A   (later: colS [64*97])
//   [ 9216,18432) xb1        -- TDM dest buffer B   (later: colQ tail)
//   [18432,27648) sA (sum)   -- later: mean[64*65], rstd[64*65]
//   [27648,36864) qA (sumsq)
// ---------------------------------------------------------------------------
__global__ __launch_bounds__(TPB)
void wingn_kernel(const float* __restrict__ x, const float* __restrict__ wgt,
                  const float* __restrict__ bia, float* __restrict__ out) {
  extern __shared__ float smem[];
  float* xb0  = smem;
  float* xb1  = smem + 9216;
  float* sA   = smem + 18432;
  float* qA   = smem + 27648;
  float* colS = smem;                  // 64*97 = 6208, overlays dead xb0
  float* colQ = smem + 6208;           // 6208, overlays xb0/xb1
  float* mA   = smem + 18432;          // 64*65 = 4160, overlays dead sA
  float* rA   = smem + 18432 + 4160;   // 4160

  const int tid = threadIdx.x;
  const int th0 = (int)(blockIdx.x >> 3) * 64;
  const int tw0 = (int)(blockIdx.x & 7) * 64;
  const int g = blockIdx.y;
  const int n = blockIdx.z;

  // stats-index base/extent: i(h) = clamp(h-15, 0, 480)  (edge pad 15/16, win 32)
  int ibh = th0 - 15; ibh = ibh < 0 ? 0 : ibh;
  int ibw = tw0 - 15; ibw = ibw < 0 ? 0 : ibw;
  int ilh = th0 + 48; ilh = ilh > 480 ? 480 : ilh;
  int ilw = tw0 + 48; ilw = ilw > 480 ? 480 : ilw;
  const int ich = ilh - ibh + 1;   // <= 64
  const int icw = ilw - ibw + 1;   // <= 64

  const int ch0 = n * 64 + g * 8;
  const float* xg0 = x + ((size_t)ch0 << 18) + (size_t)ibh * 512 + ibw;

  // ---- Pass A: per-pixel sum / sum-of-squares over the 8 group channels ----
#if USE_TDM
  const bool issuer = (tid < 32);     // wave 0 drives the TDM
  const int td0 = 512 - ibw;          // remaining cols -> OOB cols read as 0
  const int td1 = 512 - ibh;          // remaining rows -> OOB rows read as 0
  const unsigned l0 = (unsigned)(unsigned long long)(uintptr_t)xb0;
  const unsigned l1 = (unsigned)(unsigned long long)(uintptr_t)xb1;
  if (issuer) tdm_load_tile96(xg0, l0, td0, td1);
  for (int c = 0; c < 8; ++c) {
    const float* xb = (c & 1) ? xb1 : xb0;
    if (issuer) {
      if (c < 7) {  // prefetch next channel into the other buffer, keep 1 in flight
        tdm_load_tile96(xg0 + ((size_t)(c + 1) << 18), (c & 1) ? l0 : l1, td0, td1);
        wg_wait_tensorcnt1();
      } else {
        wg_wait_tensorcnt0();
      }
    }
    __syncthreads();
    if (c == 0) {
      for (int p = tid; p < 9216; p += TPB) { float v = xb[p]; sA[p] = v; qA[p] = v * v; }
    } else {
      for (int p = tid; p < 9216; p += TPB) { float v = xb[p]; sA[p] += v; qA[p] += v * v; }
    }
    __syncthreads();
  }
#else
  for (int p = tid; p < 9216; p += TPB) {
    const int r = p / 96, w2 = p - r * 96;
    const int gr = ibh + r, gc = ibw + w2;
    float aS = 0.f, aQ = 0.f;
    if (gr < 512 && gc < 512) {
      const float* px = x + ((size_t)ch0 << 18) + (size_t)gr * 512 + gc;
      for (int c = 0; c < 8; ++c) { float v = px[(size_t)c << 18]; aS += v; aQ += v * v; }
    }
    sA[p] = aS; qA[p] = aQ;
  }
  __syncthreads();
#endif

  // ---- Pass B: vertical 32-row sliding sums (thread per column) ----
  if (tid < 96) {
    const int w2 = tid;
    float aS = 0.f, aQ = 0.f;
    for (int k = 0; k < 32; ++k) { aS += sA[k * 96 + w2]; aQ += qA[k * 96 + w2]; }
    colS[w2] = aS; colQ[w2] = aQ;
    for (int i = 1; i < ich; ++i) {
      aS += sA[(i + 31) * 96 + w2] - sA[(i - 1) * 96 + w2];
      aQ += qA[(i + 31) * 96 + w2] - qA[(i - 1) * 96 + w2];
      colS[i * 97 + w2] = aS; colQ[i * 97 + w2] = aQ;
    }
  }
  __syncthreads();

  // ---- Pass C: horizontal 32-col sliding sums -> mean / rstd ----
  if (tid < ich) {
    const int i = tid;
    float aS = 0.f, aQ = 0.f;
    for (int k = 0; k < 32; ++k) { aS += colS[i * 97 + k]; aQ += colQ[i * 97 + k]; }
    const float inv_n = 1.0f / 8192.0f;   // 32*32*8
    for (int j = 0; j < icw; ++j) {
      if (j) {
        aS += colS[i * 97 + j + 31] - colS[i * 97 + j - 1];
        aQ += colQ[i * 97 + j + 31] - colQ[i * 97 + j - 1];
      }
      const float m   = aS * inv_n;
      const float var = (aQ - aS * aS * inv_n) * inv_n;   // biased, matches ref
      mA[i * 65 + j] = m;
      rA[i * 65 + j] = rsqrtf(var + 1e-5f);
    }
  }
  __syncthreads();

  // ---- Pass D: normalize + affine, b128 loads, NT b128 stores ----
  for (int c = 0; c < 8; ++c) {
    const float wv = wgt[g * 8 + c];
    const float bv = bia[g * 8 + c];
    const float* xp = x + ((size_t)(ch0 + c) << 18);
    float* op = out + ((size_t)(ch0 + c) << 18);
    for (int q = tid; q < 1024; q += TPB) {      // 1024 float4 per 64x64 tile
      const int lr  = q >> 4;
      const int lcb = (q & 15) << 2;
      const int oh  = th0 + lr;
      const int owb = tw0 + lcb;
      int cit = oh - 15; cit = cit < 0 ? 0 : (cit > 480 ? 480 : cit);
      const int ci = cit - ibh;
      const f32x4 xv = *reinterpret_cast<const f32x4*>(xp + (size_t)oh * 512 + owb);
      f32x4 ov;
#pragma unroll
      for (int e = 0; e < 4; ++e) {
        int cjt = owb + e - 15; cjt = cjt < 0 ? 0 : (cjt > 480 ? 480 : cjt);
        const int cj = cjt - ibw;
        const float m  = mA[ci * 65 + cj];
        const float rs = rA[ci * 65 + cj];
        ov[e] = (xv[e] - m) * rs * wv + bv;
      }
      __builtin_nontemporal_store(ov,
          reinterpret_cast<f32x4*>(op + (size_t)oh * 512 + owb));
    }
  }
}

// ---------------------------------------------------------------------------
extern "C" void kernel_launch(void* const* d_in, const int* in_sizes, int n_in,
                              void* d_out, int out_size, void* d_ws, size_t ws_size,
                              hipStream_t stream) {
  (void)in_sizes; (void)n_in; (void)out_size; (void)d_ws; (void)ws_size;
  const float* x = (const float*)d_in[0];
  const float* w = (const float*)d_in[1];
  const float* b = (const float*)d_in[2];
  float* out = (float*)d_out;
  dim3 grid(64, 8, 4);   // 8x8 tiles of 64x64, G=8 groups, N=4
  dim3 block(TPB);
  size_t shmem = 36864 * sizeof(float);   // 144 KB dynamic LDS
  wingn_kernel<<<grid, block, shmem, stream>>>(x, w, b, out);
}